// SRNModule_48026324303943
// MI455X (gfx1250) — compile-verified
//
#include <hip/hip_runtime.h>

typedef __attribute__((ext_vector_type(2))) float v2f;
typedef __attribute__((ext_vector_type(8))) float v8f;

#define B_  4
#define N_  4096
#define K_  32
#define C_  64
#define D1_ 64
#define D2_ 64
#define DH_ 128

// LDS pitches (in floats, kept even so float2 loads stay 8B-aligned)
#define AP 66     // rel_feat tile pitch (64 cols + pad)
#define FP 130    // fuse tile pitch (128 cols + pad)
#define XP 8      // rel_xyz tile pitch (6 cols padded to 8)

__launch_bounds__(128, 2)
__global__ void srn_main(const float* __restrict__ xyz,
                         const float* __restrict__ features,
                         const int*   __restrict__ idx,
                         const float* __restrict__ W_gu, const float* __restrict__ s_gu, const float* __restrict__ b_gu,
                         const float* __restrict__ W_gv, const float* __restrict__ s_gv, const float* __restrict__ b_gv,
                         const float* __restrict__ W_h,  const float* __restrict__ s_h,  const float* __restrict__ b_h,
                         const float* __restrict__ W_f,  const float* __restrict__ s_f,  const float* __restrict__ b_f,
                         float* __restrict__ out)
{
    __shared__ float sA[64 * AP];        // rel_feat 64x64
    __shared__ float sX[64 * XP];        // rel_xyz 64x8 (K padded)
    __shared__ float sFuse[64 * FP];     // [gu | gv] 64x128
    __shared__ float sBgu[4][2 * AP];    // W_gu^T padded to K=8, stored as (k-pair, 2n+parity)
    __shared__ float sPool[2 * DH_];     // per-point column sums of h
    __shared__ int   sIdx[64];

    const int tid  = threadIdx.x;
    const int lane = tid & 31;
    const int w    = tid >> 5;          // wave id 0..3
    const int half = lane >> 4;         // lane[4]
    const int l16  = lane & 15;

    const int blk = blockIdx.x;
    const int b   = blk / (N_ / 2);
    const int n0  = (blk % (N_ / 2)) * 2;   // first of the 2 points this block owns

    // ---- stage tiny K-padded W_gu^T; zero pool ----
    for (int e = tid; e < 8 * D1_; e += 128) {
        int n = e & 63, k = e >> 6;
        sBgu[k >> 1][2 * n + (k & 1)] = (k < 6) ? W_gu[n * 6 + k] : 0.f;
    }
    for (int e = tid; e < 2 * DH_; e += 128) sPool[e] = 0.f;
    if (tid < 64) {
        int p = tid >> 5, k = tid & 31;
        sIdx[tid] = idx[(n0 + p) * K_ + k];
    }
    __syncthreads();

    // ---- build rel_xyz tile (64 rows x 8) ----
    for (int e = tid; e < 64 * 8; e += 128) {
        int r = e >> 3, c = e & 7;
        int p = r >> 5;
        int j = sIdx[r];
        float v;
        if (c < 3)      v = xyz[((size_t)b * N_ + (n0 + p)) * 3 + c];
        else if (c < 6) v = xyz[((size_t)b * N_ + j) * 3 + (c - 3)];
        else            v = 0.f;
        sX[r * XP + c] = v;
    }
    // ---- build rel_feat tile (64 rows x 64): feats[b][c][j] + feats[b][c][n] ----
    for (int e = tid; e < 64 * 64; e += 128) {
        int r = e & 63, c = e >> 6;               // r inner: gathers stay within one feature row
        int p = r >> 5;
        int j = sIdx[r];
        const float* fb = features + ((size_t)b * C_ + c) * N_;
        sA[r * AP + c] = fb[j] + fb[n0 + p];
    }
    __syncthreads();

    const int mbase = w * 16;          // this wave's M-tile rows
    const int arow  = mbase + l16;     // A-fragment row for this lane
    const int point = w >> 1;          // waves 0,1 -> point 0; waves 2,3 -> point 1

    // ================= gu : rel_xyz(64x8) @ Wgu^T(8x64) =================
    for (int nt = 0; nt < 4; ++nt) {
        const int nb = nt * 16 + l16;
        v8f acc = {};
#pragma unroll
        for (int k0 = 0; k0 < 8; k0 += 4) {
            v2f a  = *(const v2f*)&sX[arow * XP + k0 + 2 * half];
            v2f bb = *(const v2f*)&sBgu[(k0 >> 1) + half][2 * nb];
            acc = __builtin_amdgcn_wmma_f32_16x16x4_f32(false, a, false, bb,
                                                        (short)0, acc, false, false);
        }
        const float sc = s_gu[nb], bi = b_gu[nb];
#pragma unroll
        for (int v = 0; v < 8; ++v) {
            float x = acc[v] * sc + bi;
            sFuse[(mbase + v + 8 * half) * FP + nb] = fmaxf(x, 0.f);
        }
    }
    // ================= gv : rel_feat(64x64) @ Wgv^T(64x64) ==============
    for (int nt = 0; nt < 4; ++nt) {
        const int nb = nt * 16 + l16;
        const float* wp = W_gv + nb * C_ + 2 * half;   // {W_gv[nb][k], W_gv[nb][k+1]} contiguous
        v8f acc = {};
#pragma unroll
        for (int k0 = 0; k0 < 64; k0 += 4) {
            v2f a  = *(const v2f*)&sA[arow * AP + k0 + 2 * half];
            v2f bb = *(const v2f*)(wp + k0);
            acc = __builtin_amdgcn_wmma_f32_16x16x4_f32(false, a, false, bb,
                                                        (short)0, acc, false, false);
        }
        const float sc = s_gv[nb], bi = b_gv[nb];
#pragma unroll
        for (int v = 0; v < 8; ++v) {
            float x = acc[v] * sc + bi;
            sFuse[(mbase + v + 8 * half) * FP + 64 + nb] = fmaxf(x, 0.f);
        }
    }
    __syncthreads();

    // ================= h : fuse(64x128) @ Wh^T(128x128), pooled over K ==
    for (int nt = 0; nt < 8; ++nt) {
        const int nb = nt * 16 + l16;
        const float* wp = W_h + nb * DH_ + 2 * half;
        v8f acc = {};
#pragma unroll
        for (int k0 = 0; k0 < 128; k0 += 4) {
            v2f a  = *(const v2f*)&sFuse[arow * FP + k0 + 2 * half];
            v2f bb = *(const v2f*)(wp + k0);
            acc = __builtin_amdgcn_wmma_f32_16x16x4_f32(false, a, false, bb,
                                                        (short)0, acc, false, false);
        }
        const float sc = s_h[nb], bi = b_h[nb];
        float ps = 0.f;
#pragma unroll
        for (int v = 0; v < 8; ++v) {
            float x = acc[v] * sc + bi;
            ps += fmaxf(x, 0.f);                 // sum over this lane's 8 rows (all same point)
        }
        atomicAdd(&sPool[point * DH_ + nb], ps); // 4 partials per (point,col): 2 halves x 2 waves
    }
    __syncthreads();

    // ================= final: mean over K, 1x128 @ 128x64, + features ===
    {
        const int p = tid >> 6, c = tid & 63;
        const float* wrow = W_f + c * DH_;
        const float* pool = sPool + p * DH_;
        float acc = 0.f;
#pragma unroll 4
        for (int d = 0; d < DH_; ++d) acc += pool[d] * wrow[d];
        acc *= (1.f / 32.f);                     // mean over K (linear -> fold after dot)
        float x = fmaxf(acc * s_f[c] + b_f[c], 0.f);
        const size_t o = ((size_t)b * C_ + c) * N_ + (n0 + p);
        out[o] = x + features[o];
    }
}

extern "C" void kernel_launch(void* const* d_in, const int* in_sizes, int n_in,
                              void* d_out, int out_size, void* d_ws, size_t ws_size,
                              hipStream_t stream) {
    const float* xyz      = (const float*)d_in[0];
    const float* features = (const float*)d_in[1];
    const int*   idx      = (const int*)  d_in[2];
    const float* W_gu = (const float*)d_in[3];
    const float* s_gu = (const float*)d_in[4];
    const float* b_gu = (const float*)d_in[5];
    const float* W_gv = (const float*)d_in[6];
    const float* s_gv = (const float*)d_in[7];
    const float* b_gv = (const float*)d_in[8];
    const float* W_h  = (const float*)d_in[9];
    const float* s_h  = (const float*)d_in[10];
    const float* b_h  = (const float*)d_in[11];
    const float* W_f  = (const float*)d_in[12];
    const float* s_f  = (const float*)d_in[13];
    const float* b_f  = (const float*)d_in[14];
    float* out = (float*)d_out;

    // tuple output: xyz passthrough first, then (B,C,N) feature output
    hipMemcpyAsync(out, xyz, (size_t)B_ * N_ * 3 * sizeof(float),
                   hipMemcpyDeviceToDevice, stream);

    dim3 grid(B_ * (N_ / 2));
    srn_main<<<grid, 128, 0, stream>>>(xyz, features, idx,
                                       W_gu, s_gu, b_gu,
                                       W_gv, s_gv, b_gv,
                                       W_h,  s_h,  b_h,
                                       W_f,  s_f,  b_f,
                                       out + (size_t)B_ * N_ * 3);
}